// PeriodicPrimitives2D_27195732918601
// MI455X (gfx1250) — compile-verified
//
#include <hip/hip_runtime.h>

// ---------------------------------------------------------------------------
// PeriodicPrimitives2D for MI455X (gfx1250, wave32).
// Compute-bound on the transcendental pipe (~1.2G v_cos/v_exp total);
// memory traffic is ~1.5MB -> irrelevant at 23.3 TB/s.
// Strategy:
//   * prep kernel folds rotation cos/sin + freq int->float into a packed
//     28-float per-gaussian record in d_ws (slot [9] is a 0.0f pad that the
//     B-operand reads for lanes N>=3 -> no divergent selects).
//   * main kernel double-buffers 256-gaussian tiles (2 x 28KB LDS) using
//     GLOBAL_LOAD_ASYNC_TO_LDS_B128 (ASYNCcnt) so the DMA for tile t+1
//     overlaps the ~10K cycles of compute on tile t.
//   * v_cos_f32 computes cos(2*pi*x) natively -> wave term = 1 mul + 1 cos.
//   * the (N x G) @ (G x 3) contraction is fused into the loop with
//     V_WMMA_F32_16X16X4_F32: each wave owns 16 points, each step consumes
//     4 gaussians; colors (zero-padded to 16 cols) form the B operand.
// ---------------------------------------------------------------------------

typedef __attribute__((ext_vector_type(2))) float v2f;
typedef __attribute__((ext_vector_type(8))) float v8f;

// Exact parameter types for the async-LDS builtin (from hipcc diagnostics:
// "__attribute__((__vector_size__(4 * sizeof(int)))) int *").
typedef int v4i_vs __attribute__((vector_size(4 * sizeof(int))));
typedef v4i_vs __attribute__((address_space(1)))* g128_t;  // global
typedef v4i_vs __attribute__((address_space(3)))* l128_t;  // LDS

constexpr int N_PTS  = 65536;
constexpr int G      = 2048;
constexpr int KFREQ  = 8;
constexpr int STRIDE = 28;   // floats per packed gaussian record
constexpr int TILE   = 256;  // gaussians staged in LDS per tile (28 KB)
constexpr int NTILES = G / TILE;
constexpr int NCPY   = (TILE * STRIDE / 4) / 256;  // float4s per thread = 7
constexpr float FREQ_SCALE = 1.0f;  // MAX_FREQ / NUM_TOTAL_FREQ = 128/128

// Record layout (floats):
// [0]=px [1]=py [2]=cos(th) [3]=sin(th) [4]=sx [5]=sy
// [6..8]=rgb [9]=0.0f pad [10+2k]=coeff_k [11+2k]=freq_k (already float)

#if defined(__gfx1250__) && \
    __has_builtin(__builtin_amdgcn_global_load_async_to_lds_b128) && \
    __has_builtin(__builtin_amdgcn_s_wait_asynccnt)
#define ASYNC_LDS 1
#else
#define ASYNC_LDS 0
#endif

static __device__ __forceinline__ float fast_cos2pi(float x) {
#if __has_builtin(__builtin_amdgcn_cosf)
  // v_cos_f32: D = cos(2*pi*S0); valid for |S0| <= 256.  Here |freq*rx| <
  // 128*sqrt(2) ~ 181, in range.
  return __builtin_amdgcn_cosf(x);
#else
  return __cosf(6.28318530717958647692f * x);
#endif
}

__global__ void prep_kernel(const float* __restrict__ colors,
                            const float* __restrict__ pos,
                            const float* __restrict__ scales,
                            const float* __restrict__ rots,
                            const float* __restrict__ coeffs,
                            const int*   __restrict__ idxs,
                            float*       __restrict__ ws) {
  int g = blockIdx.x * blockDim.x + threadIdx.x;
  if (g >= G) return;
  float th = rots[g];
  float* o = ws + (size_t)g * STRIDE;
  o[0] = pos[2 * g];
  o[1] = pos[2 * g + 1];
  o[2] = __cosf(th);
  o[3] = __sinf(th);
  o[4] = scales[2 * g];
  o[5] = scales[2 * g + 1];
  o[6] = colors[3 * g];
  o[7] = colors[3 * g + 1];
  o[8] = colors[3 * g + 2];
  o[9] = 0.0f;  // read by B-operand lanes with N >= 3
#pragma unroll
  for (int k = 0; k < KFREQ; ++k) {
    o[10 + 2 * k] = coeffs[g * KFREQ + k];
    o[11 + 2 * k] = (float)idxs[g * KFREQ + k] * FREQ_SCALE;
  }
}

// Kick off the copy of one 28KB tile into an LDS buffer (no wait).
static __device__ __forceinline__ void stage_tile(const float4* __restrict__ src,
                                                  float4* __restrict__ dst) {
#if ASYNC_LDS
#pragma unroll
  for (int i = 0; i < NCPY; ++i) {
    __builtin_amdgcn_global_load_async_to_lds_b128(
        (g128_t)(src + threadIdx.x + i * 256),
        (l128_t)(dst + threadIdx.x + i * 256),
        /*offset=*/0, /*cpol=*/0);
  }
#else
#pragma unroll
  for (int i = 0; i < NCPY; ++i)
    dst[threadIdx.x + i * 256] = src[threadIdx.x + i * 256];
#endif
}

static __device__ __forceinline__ void stage_wait() {
#if ASYNC_LDS
  __builtin_amdgcn_s_wait_asynccnt(0);  // my async LDS writes have landed
#endif
}

// Weight of one (point, gaussian) pair; gp points into LDS.
static __device__ __forceinline__ float eval_w(const float* __restrict__ gp,
                                               float px, float py) {
  float dx = px - gp[0];
  float dy = py - gp[1];
  float c = gp[2], s = gp[3];
  float rx = fmaf(c, dx, s * dy);    //  c*dx + s*dy
  float ry = fmaf(c, dy, -s * dx);   // -s*dx + c*dy
  float a = rx * gp[4];
  float b = ry * gp[5];
  float env = __expf(-0.5f * fmaf(a, a, b * b));
  float wave = 0.0f;
#pragma unroll
  for (int k = 0; k < KFREQ; ++k) {
    float coef = gp[10 + 2 * k];
    float freq = gp[11 + 2 * k];
    wave = fmaf(coef, fast_cos2pi(freq * rx), wave);  // cos(2*pi*freq*rx)
  }
  return env * wave;
}

__launch_bounds__(256)
__global__ void main_kernel(const float* __restrict__ x,
                            const float* __restrict__ ws,
                            float*       __restrict__ out) {
  __shared__ float lds[2][TILE * STRIDE];  // 2 x 28 KB double buffer

  const int lane = threadIdx.x & 31;   // wave32
  const int wave = threadIdx.x >> 5;   // 8 waves / block
  const int m    = lane & 15;          // A/B/D column-in-wave index
  const int half = lane >> 4;          // 0 -> K slots {0,1}, 1 -> {2,3}
  const int pbase = blockIdx.x * 128 + wave * 16;
  const int p     = pbase + m;
  // Lane-constant color slot: rgb for N<3, the 0.0f pad for N>=3.
  const int coff = (m < 3) ? (6 + m) : 9;

  const float px = x[2 * p];
  const float py = x[2 * p + 1];

  v8f acc = {};  // 16x16 f32 accumulator (only N=0..2 meaningful)

  // Prologue: stage tile 0.
  stage_tile((const float4*)ws, (float4*)lds[0]);
  stage_wait();
  __syncthreads();

  for (int t = 0; t < NTILES; ++t) {
    const float* cur = lds[t & 1];
    // Overlap: DMA tile t+1 into the other buffer while computing tile t.
    // That buffer was last read before the barrier ending iteration t-1.
    if (t + 1 < NTILES)
      stage_tile((const float4*)(ws + (size_t)(t + 1) * TILE * STRIDE),
                 (float4*)lds[(t + 1) & 1]);

    for (int g0 = 0; g0 < TILE; g0 += 4) {
      // A operand, 16x4 f32: VGPR0 holds K = 2*half, VGPR1 holds K = 2*half+1
      const float* ga = cur + (g0 + 2 * half) * STRIDE;
      const float* gb = ga + STRIDE;
      float wa = eval_w(ga, px, py);
      float wb = eval_w(gb, px, py);

      // B operand, 4x16 f32 (colors; lanes N>=3 read the 0.0f pad slot).
      float b0 = ga[coff];
      float b1 = gb[coff];

      v2f A; A.x = wa; A.y = wb;
      v2f B; B.x = b0; B.y = b1;
      acc = __builtin_amdgcn_wmma_f32_16x16x4_f32(
          /*neg_a=*/false, A, /*neg_b=*/false, B,
          /*c_mod=*/(short)0, acc, /*reuse_a=*/false, /*reuse_b=*/false);
    }

    stage_wait();      // my writes for tile t+1 are in LDS
    __syncthreads();   // everyone's writes visible; readers of tile t done
  }

  // D layout: VGPR r -> row M = r + 8*half, col N = m.  Emit N < 3 only.
  if (m < 3) {
#pragma unroll
    for (int r = 0; r < 8; ++r) {
      int row = pbase + r + 8 * half;
      out[row * 3 + m] = acc[r];
    }
  }
}

extern "C" void kernel_launch(void* const* d_in, const int* in_sizes, int n_in,
                              void* d_out, int out_size, void* d_ws, size_t ws_size,
                              hipStream_t stream) {
  const float* x      = (const float*)d_in[0];
  const float* colors = (const float*)d_in[1];
  const float* pos    = (const float*)d_in[2];
  const float* scales = (const float*)d_in[3];
  const float* rots   = (const float*)d_in[4];
  const float* coeffs = (const float*)d_in[5];
  const int*   idxs   = (const int*)d_in[6];

  float* ws  = (float*)d_ws;   // G * STRIDE * 4 = 229 KB of scratch
  float* out = (float*)d_out;

  prep_kernel<<<(G + 255) / 256, 256, 0, stream>>>(colors, pos, scales, rots,
                                                   coeffs, idxs, ws);
  main_kernel<<<N_PTS / 128, 256, 0, stream>>>(x, ws, out);
}